// MaterialMLP_20057497272431
// MI455X (gfx1250) — compile-verified
//
#include <hip/hip_runtime.h>

// MI455X / gfx1250, wave32. Streaming BRDF shade: one wave per pixel,
// lane l handles samples l and l+32. Cross-lane reduction via
// V_WMMA_F32_16X16X4_F32 with B=ones, finished by a single ds_swizzle
// SWAPX16 (16x16 f32 C/D layout: vgpr r = row r in lanes
// 0-15, row r+8 in lanes 16-31).

typedef __attribute__((ext_vector_type(2))) float v2f;
typedef __attribute__((ext_vector_type(8))) float v8f;

#define TINYV 1e-6f
#define NSAMP 64
#define NPIX  (256 * 512)

__device__ __forceinline__ float clamp01(float x) {
  return __builtin_fminf(__builtin_fmaxf(x, 0.0f), 1.0f);
}
__device__ __forceinline__ float rcpf(float x) {
  return __builtin_amdgcn_rcpf(x);
}
// lane <-> lane^16 swap, group-of-32 ds_swizzle imm: and=0x1f, or=0, xor=0x10
__device__ __forceinline__ float swap16(float x) {
  return __int_as_float(
      __builtin_amdgcn_ds_swizzle(__float_as_int(x), 0x401F));
}

__global__ __launch_bounds__(256) void brdf_shade_kernel(
    const float* __restrict__ normal,     // (N,3)
    const float* __restrict__ albedo,     // (N,3)
    const float* __restrict__ roughness,  // (N,1)
    const float* __restrict__ points,     // (N,3)
    const float* __restrict__ cam,        // (3,)
    const float* __restrict__ ldir,       // (N,S,3) light_dir_diff
    const float* __restrict__ dlit,       // (N,S,3) diffuse_lighting
    const float* __restrict__ hdir,       // (N,S,3) h_dir_specular
    const float* __restrict__ slit,       // (N,S,3) specular_lighting
    float* __restrict__ out)              // (N,3)
{
  const int wave = threadIdx.x >> 5;
  const int lane = threadIdx.x & 31;
  const int p = (blockIdx.x << 3) + wave;   // grid covers NPIX exactly

  // ---- per-pixel (wave-uniform) setup ----
  const float nx = normal[p * 3 + 0];
  const float ny = normal[p * 3 + 1];
  const float nz = normal[p * 3 + 2];
  const float r  = roughness[p];
  const float ppx = points[p * 3 + 0];
  const float ppy = points[p * 3 + 1];
  const float ppz = points[p * 3 + 2];
  const float dxx = cam[0] - ppx;
  const float dyy = cam[1] - ppy;
  const float dzz = cam[2] - ppz;
  const float len  = __builtin_sqrtf(dxx * dxx + dyy * dyy + dzz * dzz);
  const float invl = rcpf(__builtin_fmaxf(len, 1e-4f));
  const float vx = dxx * invl, vy = dyy * invl, vz = dzz * invl;

  const float ndv = clamp01(nx * vx + ny * vy + nz * vz);
  const float k   = (r + 1.0f) * (r + 1.0f) * 0.125f;
  const float omk = 1.0f - k;
  const float g1v = ndv * rcpf(__builtin_fmaxf(ndv * omk + k, TINYV));

  const float invS = 1.0f / (float)NSAMP;
  // diffuse weight: (a/pi) * 2*pi / S == a * 2 / S
  const float a2x = albedo[p * 3 + 0] * (2.0f * invS);
  const float a2y = albedo[p * 3 + 1] * (2.0f * invS);
  const float a2z = albedo[p * 3 + 2] * (2.0f * invS);

  // ---- per-sample streaming: 2 samples per lane ----
  float cr[2], cg[2], cb[2];
  const int base = p * (NSAMP * 3);
#pragma unroll
  for (int j = 0; j < 2; ++j) {
    const int o = base + (lane + j * 32) * 3;
    // 4 coalesced b96 streams (384 contiguous bytes per wave per load)
    const float lx = ldir[o + 0], ly = ldir[o + 1], lz = ldir[o + 2];
    const float er = dlit[o + 0], eg = dlit[o + 1], eb = dlit[o + 2];
    const float hx = hdir[o + 0], hy = hdir[o + 1], hz = hdir[o + 2];
    const float sr = slit[o + 0], sg = slit[o + 1], sb = slit[o + 2];

    // diffuse
    const float ndl_d = clamp01(nx * lx + ny * ly + nz * lz);
    float ccr = er * (a2x * ndl_d);
    float ccg = eg * (a2y * ndl_d);
    float ccb = eb * (a2z * ndl_d);

    // specular
    const float vdh = clamp01(hx * vx + hy * vy + hz * vz);
    const float t2  = 2.0f * vdh;
    const float qx = t2 * hx - vx, qy = t2 * hy - vy, qz = t2 * hz - vz;
    const float ndl = clamp01(nx * qx + ny * qy + nz * qz);
    const float ndh = clamp01(nx * hx + ny * hy + nz * hz);
    const float fr  = 0.04f +
        0.96f * __builtin_exp2f((-5.55472f * vdh - 6.98316f) * vdh);
    const float g1l  = ndl * rcpf(__builtin_fmaxf(ndl * omk + k, TINYV));
    const float brdf = fr * (g1l * g1v) *
        rcpf(__builtin_fmaxf(4.0f * ndl * ndv, TINYV));
    const float w = brdf * ndl * 4.0f * vdh *
        rcpf(__builtin_fmaxf(ndh, TINYV)) * invS;
    ccr += sr * w;
    ccg += sg * w;
    ccb += sb * w;

    cr[j] = ccr; cg[j] = ccg; cb[j] = ccb;
  }

  // ---- wave reduction via f32 WMMA (CDNA5) ----
  // B = all ones => D[m,n] = sum_k A[m,k]  (row sum, column-constant).
  // Row m sums lanes {m, m+16} x regs {a0, a1}.
  const v2f ones = {1.0f, 1.0f};
  const v8f zc   = {};
  v2f aR; aR.x = cr[0]; aR.y = cr[1];
  v2f aG; aG.x = cg[0]; aG.y = cg[1];
  v2f aB; aB.x = cb[0]; aB.y = cb[1];

  v8f dR = __builtin_amdgcn_wmma_f32_16x16x4_f32(false, aR, false, ones,
                                                 (short)0, zc, false, false);
  v8f dG = __builtin_amdgcn_wmma_f32_16x16x4_f32(false, aG, false, ones,
                                                 (short)0, zc, false, false);
  v8f dB = __builtin_amdgcn_wmma_f32_16x16x4_f32(false, aB, false, ones,
                                                 (short)0, zc, false, false);

  // Per-lane sum of the 8 D vgprs: lanes 0-15 -> sum rows 0-7,
  // lanes 16-31 -> sum rows 8-15 (documented C/D layout). One SWAPX16
  // swizzle + add completes the 32-lane reduction.
  float sR = dR[0] + dR[1] + dR[2] + dR[3] + dR[4] + dR[5] + dR[6] + dR[7];
  float sG = dG[0] + dG[1] + dG[2] + dG[3] + dG[4] + dG[5] + dG[6] + dG[7];
  float sB = dB[0] + dB[1] + dB[2] + dB[3] + dB[4] + dB[5] + dB[6] + dB[7];
  sR += swap16(sR);
  sG += swap16(sG);
  sB += swap16(sB);

  if (lane == 0) {
    out[p * 3 + 0] = sR;
    out[p * 3 + 1] = sG;
    out[p * 3 + 2] = sB;
  }
}

extern "C" void kernel_launch(void* const* d_in, const int* in_sizes, int n_in,
                              void* d_out, int out_size, void* d_ws,
                              size_t ws_size, hipStream_t stream) {
  (void)in_sizes; (void)n_in; (void)out_size; (void)d_ws; (void)ws_size;
  const float* normal    = (const float*)d_in[0];
  const float* albedo    = (const float*)d_in[1];
  const float* roughness = (const float*)d_in[2];
  const float* points    = (const float*)d_in[3];
  const float* cam       = (const float*)d_in[4];
  const float* ldir      = (const float*)d_in[5];
  const float* dlit      = (const float*)d_in[6];
  const float* hdir      = (const float*)d_in[7];
  const float* slit      = (const float*)d_in[8];
  float* out = (float*)d_out;

  // 8 waves (8 pixels) per 256-thread block; grid covers all pixels exactly,
  // so EXEC is full everywhere (WMMA requirement).
  brdf_shade_kernel<<<NPIX / 8, 256, 0, stream>>>(
      normal, albedo, roughness, points, cam, ldir, dlit, hdir, slit, out);
}